// DGP_75960791597104
// MI455X (gfx1250) — compile-verified
//
#include <hip/hip_runtime.h>
#include <hip/hip_bf16.h>
#include <math.h>

// Problem constants (from reference)
#define BB   8
#define NOBS 256
#define NTGT 256
#define ZZ   16
#define RR   64
#define HH   256
#define DD   128
#define TT   512   // NOBS + NTGT

typedef float v2f __attribute__((ext_vector_type(2)));
typedef float v8f __attribute__((ext_vector_type(8)));

// fp32 WMMA: D(16x16) = A(16x4) * B(4x16) + C  -- exact fp32 on the matrix pipe
__device__ __forceinline__ v8f wmma_f32(v2f a, v2f b, v8f c) {
  return __builtin_amdgcn_wmma_f32_16x16x4_f32(false, a, false, b, (short)0, c,
                                               false, false);
}

// ---------------------------------------------------------------------------
// Layer 1: h1 = relu(concat(point_x,target_x) @ W1[z] + b1[z])
// h1 layout [B,T,Z,H]  (row m = b*T+t contiguous in H)
// ---------------------------------------------------------------------------
__global__ void k_mlp1(const float* __restrict__ px, const float* __restrict__ tx,
                       const float* __restrict__ W1, const float* __restrict__ b1,
                       float* __restrict__ h1) {
  int w    = (blockIdx.x * 256 + threadIdx.x) >> 5;  // global wave id
  int lane = threadIdx.x & 31;
  int nt = w & 15, mt = (w >> 4) & 255, z = w >> 12;
  int m0 = mt * 16, n0 = nt * 16;
  int half = lane >> 4, col = lane & 15;

  // A row owned by this lane (tile never crosses a batch or the obs/tgt split)
  int m = m0 + col;
  int bb = m >> 9;           // / T
  int t  = m & 511;          // % T
  const float* arow = (t < NOBS)
      ? (px + (((size_t)bb * NOBS + t) * ZZ + z) * RR)
      : (tx + (((size_t)bb * NTGT + (t - NOBS)) * ZZ + z) * RR);

  v8f c = {};
  for (int kb = 0; kb < RR; kb += 4) {
    v2f a, bmat;
    a.x = arow[kb + 2 * half];
    a.y = arow[kb + 2 * half + 1];
    const float* bp = W1 + ((size_t)z * RR + kb + 2 * half) * HH + n0 + col;
    bmat.x = bp[0];
    bmat.y = bp[HH];
    c = wmma_f32(a, bmat, c);
  }
  int n = n0 + col;
  float bias = b1[z * HH + n];
#pragma unroll
  for (int v = 0; v < 8; ++v) {
    int row = m0 + v + 8 * half;
    h1[((size_t)row * ZZ + z) * HH + n] = fmaxf(c[v] + bias, 0.0f);
  }
}

// ---------------------------------------------------------------------------
// Layer 2: h2 = relu(h1 @ W2[z] + b2[z]),  K = H = 256
// ---------------------------------------------------------------------------
__global__ void k_mlp2(const float* __restrict__ h1, const float* __restrict__ W2,
                       const float* __restrict__ b2, float* __restrict__ h2) {
  int w    = (blockIdx.x * 256 + threadIdx.x) >> 5;
  int lane = threadIdx.x & 31;
  int nt = w & 15, mt = (w >> 4) & 255, z = w >> 12;
  int m0 = mt * 16, n0 = nt * 16;
  int half = lane >> 4, col = lane & 15;

  const float* arow = h1 + ((size_t)(m0 + col) * ZZ + z) * HH;
  v8f c = {};
  for (int kb = 0; kb < HH; kb += 4) {
    v2f a, bmat;
    a.x = arow[kb + 2 * half];
    a.y = arow[kb + 2 * half + 1];
    const float* bp = W2 + ((size_t)z * HH + kb + 2 * half) * HH + n0 + col;
    bmat.x = bp[0];
    bmat.y = bp[HH];
    c = wmma_f32(a, bmat, c);
  }
  int n = n0 + col;
  float bias = b2[z * HH + n];
#pragma unroll
  for (int v = 0; v < 8; ++v) {
    int row = m0 + v + 8 * half;
    h2[((size_t)row * ZZ + z) * HH + n] = fmaxf(c[v] + bias, 0.0f);
  }
}

// ---------------------------------------------------------------------------
// Layer 3: f = (h2 @ W3[z] + b3[z]) * (1 + unoise/20), stored [B,Z,T,D]
// ---------------------------------------------------------------------------
__global__ void k_mlp3(const float* __restrict__ h2, const float* __restrict__ W3,
                       const float* __restrict__ b3, const float* __restrict__ un,
                       float* __restrict__ f) {
  int w    = (blockIdx.x * 256 + threadIdx.x) >> 5;
  int lane = threadIdx.x & 31;
  int nt = w & 7, mt = (w >> 3) & 255, z = w >> 11;
  int m0 = mt * 16, n0 = nt * 16;
  int half = lane >> 4, col = lane & 15;

  const float* arow = h2 + ((size_t)(m0 + col) * ZZ + z) * HH;
  v8f c = {};
  for (int kb = 0; kb < HH; kb += 4) {
    v2f a, bmat;
    a.x = arow[kb + 2 * half];
    a.y = arow[kb + 2 * half + 1];
    const float* bp = W3 + ((size_t)z * HH + kb + 2 * half) * DD + n0 + col;
    bmat.x = bp[0];
    bmat.y = bp[DD];
    c = wmma_f32(a, bmat, c);
  }
  int n = n0 + col;
  float bias = b3[z * DD + n];
#pragma unroll
  for (int v = 0; v < 8; ++v) {
    int row = m0 + v + 8 * half;
    int bb = row >> 9, t = row & 511;
    size_t idx = (((size_t)bb * ZZ + z) * TT + t) * DD + n;
    f[idx] = (c[v] + bias) * (1.0f + 0.05f * un[idx]);
  }
}

// ---------------------------------------------------------------------------
// sq[b,z,t] = sum_d f^2
// ---------------------------------------------------------------------------
__global__ void k_sq(const float* __restrict__ f, float* __restrict__ sq) {
  int i = blockIdx.x * 256 + threadIdx.x;  // 0 .. B*Z*T-1
  const float* fr = f + (size_t)i * DD;
  float s = 0.0f;
  for (int d = 0; d < DD; ++d) s += fr[d] * fr[d];
  sq[i] = s;
}

// ---------------------------------------------------------------------------
// Gram + kernel matrix: K[bz,i,j] = exp(-0.5*max(sq_i+sq_j-2*<f_i,f_j>, 0))
// plus diag output. One wave per 16x16 tile, WMMA over D=128.
// ---------------------------------------------------------------------------
__global__ void k_gram(const float* __restrict__ f, const float* __restrict__ sq,
                       float* __restrict__ Kout, float* __restrict__ diag) {
  int w    = (blockIdx.x * 256 + threadIdx.x) >> 5;
  int lane = threadIdx.x & 31;
  int jt = w & 31, it = (w >> 5) & 31, bz = w >> 10;
  int i0 = it * 16, j0 = jt * 16;
  int half = lane >> 4, col = lane & 15;

  const float* fb = f + (size_t)bz * TT * DD;
  const float* ai = fb + (size_t)(i0 + col) * DD;  // A row (lane owns row i0+col)
  const float* bj = fb + (size_t)(j0 + col) * DD;  // B col = F row j0+col

  v8f c = {};
  for (int kb = 0; kb < DD; kb += 4) {
    v2f a, bmat;
    a.x = ai[kb + 2 * half];
    a.y = ai[kb + 2 * half + 1];
    bmat.x = bj[kb + 2 * half];
    bmat.y = bj[kb + 2 * half + 1];
    c = wmma_f32(a, bmat, c);
  }
  int j = j0 + col;
  float sqj = sq[bz * TT + j];
  float* kb_out = Kout + (size_t)bz * TT * TT;
#pragma unroll
  for (int v = 0; v < 8; ++v) {
    int i = i0 + v + 8 * half;
    float d2 = sq[bz * TT + i] + sqj - 2.0f * c[v];
    float kv = expf(-0.5f * fmaxf(d2, 0.0f));
    kb_out[(size_t)i * TT + j] = kv;
    if (i == j) diag[bz * TT + i] = kv;
  }
}

// ---------------------------------------------------------------------------
// Per (b,z): Cholesky of A=K[:256,:256] in LDS (256KB, CDNA5 WGP has 320KB),
// solve V = A^{-1} kqo^T (256 RHS, thread per column, V in global ws),
// alpha = A^{-1} y (wave0 with shfl reductions), then mu/var/sigma epilogue.
// ---------------------------------------------------------------------------
__global__ void k_solve(const float* __restrict__ Kout, const float* __restrict__ diag,
                        const float* __restrict__ py, const float* __restrict__ eps,
                        float* __restrict__ Vws, float* __restrict__ o_ty,
                        float* __restrict__ o_mu, float* __restrict__ o_lv) {
  extern __shared__ float lds[];
  float* A     = lds;                 // 256*256
  float* alpha = lds + NOBS * NOBS;   // 256

  int bz = blockIdx.x;
  int bb = bz >> 4, z = bz & 15;
  int tid = threadIdx.x;

  const float* kb_base = Kout + (size_t)bz * TT * TT;

  // Load A = K[:NOBS,:NOBS] into LDS (coalesced)
  for (int idx = tid; idx < NOBS * NOBS; idx += 256) {
    int i = idx >> 8, j = idx & 255;
    A[idx] = kb_base[(size_t)i * TT + j];
  }
  __syncthreads();

  // In-place Cholesky (lower triangle)
  for (int k = 0; k < NOBS; ++k) {
    if (tid == 0) A[k * 257] = sqrtf(A[k * 257]);
    __syncthreads();
    float dk = A[k * 257];
    int r = k + 1 + tid;
    if (r < NOBS) A[r * 256 + k] /= dk;
    __syncthreads();
    int jcol = k + 1 + tid;
    if (jcol < NOBS) {
      float ljk = A[jcol * 256 + k];
      for (int i = jcol; i < NOBS; ++i)
        A[i * 256 + jcol] -= A[i * 256 + k] * ljk;
    }
    __syncthreads();
  }

  // V solve: L L^T V = kqo^T ; thread c owns RHS column c (target c)
  {
    int c = tid;
    const float* kqoRow = kb_base + (size_t)(NOBS + c) * TT;  // kqo^T[i][c]=K[NOBS+c][i]
    float* Vb = Vws + (size_t)bz * NOBS * NTGT;
    // forward: L x = rhs
    for (int i = 0; i < NOBS; ++i) {
      float acc = kqoRow[i];
      const float* Li = A + i * 256;
      for (int j = 0; j < i; ++j) acc -= Li[j] * Vb[j * 256 + c];
      Vb[i * 256 + c] = acc / Li[i];
    }
    // backward: L^T v = x
    for (int i = NOBS - 1; i >= 0; --i) {
      float acc = Vb[i * 256 + c];
      for (int j = i + 1; j < NOBS; ++j) acc -= A[j * 256 + i] * Vb[j * 256 + c];
      Vb[i * 256 + c] = acc / A[i * 257];
    }
  }

  // alpha solve (wave 0), y[i] = point_y[b, i, z]
  if (tid < 32) {
    for (int i = 0; i < NOBS; ++i) {
      float s = 0.0f;
      for (int j = tid; j < i; j += 32) s += A[i * 256 + j] * alpha[j];
#pragma unroll
      for (int off = 16; off > 0; off >>= 1) s += __shfl_xor(s, off, 32);
      if (tid == 0)
        alpha[i] = (py[((size_t)bb * NOBS + i) * ZZ + z] - s) / A[i * 257];
    }
    for (int i = NOBS - 1; i >= 0; --i) {
      float s = 0.0f;
      for (int j = i + 1 + tid; j < NOBS; j += 32) s += A[j * 256 + i] * alpha[j];
#pragma unroll
      for (int off = 16; off > 0; off >>= 1) s += __shfl_xor(s, off, 32);
      if (tid == 0) alpha[i] = (alpha[i] - s) / A[i * 257];
    }
  }
  __syncthreads();

  // Epilogue: thread t handles target t
  {
    int t = tid;
    const float* kqoRow = kb_base + (size_t)(NOBS + t) * TT;
    const float* Vb = Vws + (size_t)bz * NOBS * NTGT;
    float mu = 0.0f, quad = 0.0f;
    for (int n = 0; n < NOBS; ++n) {
      float kv = kqoRow[n];
      mu   += kv * alpha[n];
      quad += kv * Vb[n * 256 + t];
    }
    float cqq = diag[bz * TT + NOBS + t];
    float var = cqq - quad;
    float sig = sqrtf(fmaxf(var, 0.0f));
    sig = (sig < 1e-5f) ? 1e-5f : sig;
    size_t oidx = ((size_t)bb * NTGT + t) * ZZ + z;
    o_ty[oidx] = mu + sig * eps[oidx];
    o_mu[oidx] = mu;
    o_lv[oidx] = 2.0f * logf(sig);
  }
}

// ---------------------------------------------------------------------------
extern "C" void kernel_launch(void* const* d_in, const int* in_sizes, int n_in,
                              void* d_out, int out_size, void* d_ws, size_t ws_size,
                              hipStream_t stream) {
  (void)in_sizes; (void)n_in; (void)out_size; (void)ws_size;
  const float* point_y  = (const float*)d_in[0];
  const float* point_x  = (const float*)d_in[1];
  const float* target_x = (const float*)d_in[2];
  const float* W1 = (const float*)d_in[3];
  const float* b1 = (const float*)d_in[4];
  const float* W2 = (const float*)d_in[5];
  const float* b2 = (const float*)d_in[6];
  const float* W3 = (const float*)d_in[7];
  const float* b3 = (const float*)d_in[8];
  const float* un = (const float*)d_in[9];
  const float* eps = (const float*)d_in[10];

  float* out   = (float*)d_out;
  float* o_ty  = out;                        // [B,NTGT,Z]
  float* o_mu  = out + 32768;                // [B,NTGT,Z]
  float* o_lv  = out + 65536;                // [B,NTGT,Z]
  float* o_dg  = out + 98304;                // [B,Z,T]
  float* o_K   = out + 163840;               // [B,Z,T,T]

  float* ws = (float*)d_ws;
  float* h1 = ws;                  // 16,777,216 floats [B,T,Z,H]
  float* h2 = ws + 16777216;       // 16,777,216 floats
  float* f  = ws;                  // aliases h1 (h1 dead after mlp2)
  float* Vg = ws + 16777216;       // aliases h2 (h2 dead after mlp3): 8,388,608
  float* sq = ws + 33554432;       // 65,536 floats

  k_mlp1<<<8192, 256, 0, stream>>>(point_x, target_x, W1, b1, h1);
  k_mlp2<<<8192, 256, 0, stream>>>(h1, W2, b2, h2);
  k_mlp3<<<4096, 256, 0, stream>>>(h2, W3, b3, un, f);
  k_sq  <<<256,  256, 0, stream>>>(f, sq);
  k_gram<<<16384, 256, 0, stream>>>(f, sq, o_K, o_dg);
  // LDS: 256*256 (A) + 256 (alpha) floats = 263,168 bytes (fits 320KB WGP LDS)
  k_solve<<<128, 256, 263168, stream>>>(o_K, o_dg, point_y, eps, Vg,
                                        o_ty, o_mu, o_lv);
}